// LinearActorNet_515396076051
// MI455X (gfx1250) — compile-verified
//
#include <hip/hip_runtime.h>
#include <hip/hip_bf16.h>
#include <stdint.h>

#define B_SZ 2048
#define K_SZ 512
#define N_SZ 256
#define T_SZ 64
#define TILE_M 16
#define MAX_TILES (B_SZ / TILE_M + T_SZ)   // 192: each task adds at most one partial tile

typedef __attribute__((ext_vector_type(2))) float v2f;
typedef __attribute__((ext_vector_type(4))) float v4f;
typedef __attribute__((ext_vector_type(8))) float v8f;

// workspace layout (in ints)
#define WS_ORDER 0                      // [B_SZ]   sample indices sorted by task
#define WS_TT    (WS_ORDER + B_SZ)      // [MAX_TILES] tile -> task
#define WS_TS    (WS_TT + MAX_TILES)    // [MAX_TILES] tile -> start offset in order[]
#define WS_TR    (WS_TS + MAX_TILES)    // [MAX_TILES] tile -> valid rows (1..16)
#define WS_NT    (WS_TR + MAX_TILES)    // [1] number of tiles

// ---------------------------------------------------------------------------
// Kernel 1: counting-sort samples by task, build 16-row tile descriptors.
// ---------------------------------------------------------------------------
__global__ void actor_sort_by_task(const int* __restrict__ task_id,
                                   int* __restrict__ ws) {
  __shared__ int cnt[T_SZ];
  __shared__ int off[T_SZ];
  __shared__ int cur[T_SZ];
  const int tid = threadIdx.x;

  if (tid < T_SZ) cnt[tid] = 0;
  __syncthreads();
  for (int b = tid; b < B_SZ; b += blockDim.x)
    atomicAdd(&cnt[task_id[b]], 1);
  __syncthreads();

  if (tid == 0) {
    int running = 0, nt = 0;
    for (int t = 0; t < T_SZ; ++t) {
      off[t] = running;
      const int c = cnt[t];
      for (int j = 0; j < c; j += TILE_M) {
        ws[WS_TT + nt] = t;
        ws[WS_TS + nt] = running + j;
        ws[WS_TR + nt] = (c - j < TILE_M) ? (c - j) : TILE_M;
        ++nt;
      }
      running += c;
    }
    ws[WS_NT] = nt;
  }
  __syncthreads();

  if (tid < T_SZ) cur[tid] = off[tid];
  __syncthreads();
  for (int b = tid; b < B_SZ; b += blockDim.x) {
    const int t = task_id[b];
    const int pos = atomicAdd(&cur[t], 1);
    ws[WS_ORDER + pos] = b;
  }
}

// ---------------------------------------------------------------------------
// Kernel 2: one block = one 16-sample tile of a single task.
// 512 threads = 16 waves; wave w owns the 16-wide N-slice [16w, 16w+16).
// A staged via global_load_async_to_lds_b128; GEMM via v_wmma_f32_16x16x4_f32;
// then per-row log-softmax + entropy with wave32 reductions.
// ---------------------------------------------------------------------------
__global__ __launch_bounds__(512)
void actor_tile_kernel(const float* __restrict__ xs,
                       const int*   __restrict__ action,
                       const float* __restrict__ weights,
                       const float* __restrict__ biases,
                       float*       __restrict__ out,
                       const int*   __restrict__ ws) {
  __shared__ float Alds[TILE_M][K_SZ];   // 32 KB gathered xs rows
  __shared__ float Llds[TILE_M][N_SZ];   // 16 KB logits tile
  __shared__ int   samp[TILE_M];
  __shared__ int   tinfo[3];

  const int tid = threadIdx.x;
  const int bid = blockIdx.x;
  if (bid >= ws[WS_NT]) return;          // block-uniform

  if (tid == 0) {
    tinfo[0] = ws[WS_TT + bid];
    tinfo[1] = ws[WS_TS + bid];
    tinfo[2] = ws[WS_TR + bid];
  }
  __syncthreads();
  const int task  = tinfo[0];
  const int start = tinfo[1];
  const int rows  = tinfo[2];

  if (tid < TILE_M) samp[tid] = (tid < rows) ? ws[WS_ORDER + start + tid] : -1;
  __syncthreads();

  // Stage 16 gathered xs rows into LDS via the CDNA5 async copy path
  // (memory -> LDS directly, tracked with ASYNCcnt; no VGPR round-trip).
  // Padded rows get zeros via a plain ds_store under the EXEC mask.
  for (int i = tid; i < TILE_M * (K_SZ / 4); i += 512) {
    const int r  = i >> 7;               // / (K_SZ/4)
    const int c4 = i & 127;
    const int s  = samp[r];
    float* dst = &Alds[r][c4 * 4];
    if (s >= 0) {
      const uint32_t lds_addr = (uint32_t)(uintptr_t)(void*)dst;
      const uint64_t gaddr = (uint64_t)(uintptr_t)(xs + (size_t)s * K_SZ + c4 * 4);
      asm volatile("global_load_async_to_lds_b128 %0, %1, off"
                   :: "v"(lds_addr), "v"(gaddr) : "memory");
    } else {
      *((v4f*)dst) = (v4f){0.f, 0.f, 0.f, 0.f};
    }
  }
  asm volatile("s_wait_asynccnt 0x0" ::: "memory");
  __syncthreads();

  const int wave = tid >> 5;
  const int lane = tid & 31;
  const int nb   = wave * 16;            // this wave's N base (0..240)
  const int m    = lane & 15;            // A-frag row
  const int kh   = (lane >> 4) << 1;     // K half-select: 0 or 2
  const int n    = nb + (lane & 15);     // B/C-frag column

  // Column pointer into this task's weights (K x N row-major, L2-resident).
  const float* Wt = weights + (size_t)task * K_SZ * N_SZ + n;

  // Hint the K-chunks of this wave's weight columns into cache up front
  // (branch-free; removes per-iteration SALU compare/branch overhead).
  #pragma unroll
  for (int pk = 64; pk < K_SZ; pk += 64)
    __builtin_prefetch(Wt + (size_t)pk * N_SZ, 0, 0);

  v8f acc = {0.f, 0.f, 0.f, 0.f, 0.f, 0.f, 0.f, 0.f};

  #pragma unroll 4
  for (int k0 = 0; k0 < K_SZ; k0 += 4) {
    v2f a;                               // A: 16x4 fp32 frag (ds_load_b64)
    a.x = Alds[m][k0 + kh];
    a.y = Alds[m][k0 + kh + 1];

    v2f b;                               // B: 4x16 fp32 frag
    b.x = Wt[(size_t)(k0 + kh) * N_SZ];
    b.y = Wt[(size_t)(k0 + kh + 1) * N_SZ];

    // 8 args: (neg_a, A, neg_b, B, c_mod, C, reuse_a, reuse_b)
    acc = __builtin_amdgcn_wmma_f32_16x16x4_f32(
        false, a, false, b, (short)0, acc, false, false);
  }

  // Spill C tile to LDS with bias added.
  // C layout: VGPR v -> (M = v + 8*(lane>>4), N = nb + (lane&15)).
  const float bv   = biases[task * N_SZ + n];
  const int   mrow = (lane >> 4) * 8;
  #pragma unroll
  for (int v = 0; v < 8; ++v)
    Llds[mrow + v][n] = acc[v] + bv;
  __syncthreads();

  // Softmax: wave w handles tile row w (8 logits per lane, wave32 reductions).
  const int row = wave;
  const int s   = samp[row];

  float l[8];
  float lmax = -1e30f;
  #pragma unroll
  for (int i = 0; i < 8; ++i) {
    l[i] = Llds[row][lane + i * 32];
    lmax = fmaxf(lmax, l[i]);
  }
  for (int o = 16; o; o >>= 1) lmax = fmaxf(lmax, __shfl_xor(lmax, o, 32));

  float sum = 0.f;
  #pragma unroll
  for (int i = 0; i < 8; ++i) sum += __expf(l[i] - lmax);
  for (int o = 16; o; o >>= 1) sum += __shfl_xor(sum, o, 32);

  const float logZ = lmax + __logf(sum);

  float ent = 0.f;
  #pragma unroll
  for (int i = 0; i < 8; ++i) {
    const float lp = l[i] - logZ;
    ent -= __expf(lp) * lp;
  }
  for (int o = 16; o; o >>= 1) ent += __shfl_xor(ent, o, 32);

  if (s >= 0 && lane == 0) {
    const int a = action[s];
    out[s]            = (float)a;                  // action passthrough
    out[B_SZ + s]     = Llds[row][a] - logZ;       // log_prob
    out[2 * B_SZ + s] = ent;                       // entropy
  }
}

// ---------------------------------------------------------------------------
extern "C" void kernel_launch(void* const* d_in, const int* in_sizes, int n_in,
                              void* d_out, int out_size, void* d_ws, size_t ws_size,
                              hipStream_t stream) {
  const float* xs      = (const float*)d_in[0];
  const int*   task_id = (const int*)  d_in[1];
  const int*   action  = (const int*)  d_in[2];
  const float* weights = (const float*)d_in[3];
  const float* biases  = (const float*)d_in[4];
  float*       out     = (float*)d_out;
  int*         ws      = (int*)d_ws;

  actor_sort_by_task<<<1, 256, 0, stream>>>(task_id, ws);
  actor_tile_kernel<<<MAX_TILES, 512, 0, stream>>>(xs, action, weights, biases, out, ws);
}